// Sine_36782099923394
// MI455X (gfx1250) — compile-verified
//
#include <hip/hip_runtime.h>
#include <math.h>

// ---------------------------------------------------------------------------
// Sine oscillator: out[b,t,c] = sin(theta0[b,c] + 2*pi*dt*cumsum_t(freq)) * amp
// freq = (sr/2)*af[...,1], amp = af[...,0].
// 3-pass hierarchical scan; intra-tile scan via V_WMMA_F32_16X16X4_F32 against
// a constant lower-triangular ones matrix. Phase tracked in revolutions ->
// v_fract_f32 + v_sin_f32 (hardware sin), no libm range reduction.
// ---------------------------------------------------------------------------

typedef __attribute__((ext_vector_type(2))) float v2f;
typedef __attribute__((ext_vector_type(8))) float v8f;

#define T_TOTAL 44100
#define NB      16
#define NC      64
#define TCH     448          // chunk length (28 tiles of 16)
#define NTILES  28
#define NCHUNK  99           // ceil(44100/448): 98 full + tail of 196
#define WPB     8            // waves per 256-thread block
#define ROWF    (NC * 2)     // floats per timestep row in af

// ---------------------------------------------------------------------------
// Pass 1: per-chunk sums of af[...,1] for 16 channels per wave (2 time-phases).
// ws layout: ws[chunk][b][c]  (coalesced along c)
// ---------------------------------------------------------------------------
__global__ void osc_partial_sums(const float* __restrict__ af,
                                 float* __restrict__ ws) {
    int w     = blockIdx.x * WPB + (threadIdx.x >> 5);
    int lane  = threadIdx.x & 31;
    int chunk = w % NCHUNK;
    int rem   = w / NCHUNK;
    int cblk  = rem & 3;
    int b     = rem >> 2;
    int n     = lane & 15;
    int th    = lane >> 4;
    int c     = cblk * 16 + n;
    int t0    = chunk * TCH;

    int tend = (t0 + TCH <= T_TOTAL) ? TCH : (T_TOTAL - t0);
    const float* fp = af + (b * T_TOTAL + t0) * ROWF + c * 2 + 1;

    float s = 0.f;
    for (int i = th; i < tend; i += 2)
        s += __builtin_nontemporal_load(&fp[i * ROWF]);

    // fold the two half-wave partials (lane <-> lane^16)
    int o = __builtin_amdgcn_ds_bpermute((lane ^ 16) << 2, __float_as_int(s));
    s += __int_as_float(o);
    if (lane < 16) ws[(chunk * NB + b) * NC + c] = s;
}

// ---------------------------------------------------------------------------
// Pass 2: exclusive scan of the 99 chunk sums per (b,c). 1024 threads total.
// ---------------------------------------------------------------------------
__global__ void osc_scan_partials(float* __restrict__ ws) {
    int idx = blockIdx.x * blockDim.x + threadIdx.x;   // b*NC + c
    if (idx >= NB * NC) return;
    float run = 0.f;
    for (int k = 0; k < NCHUNK; ++k) {
        float v = ws[k * NB * NC + idx];
        ws[k * NB * NC + idx] = run;
        run += v;
    }
}

// ---------------------------------------------------------------------------
// Pass 3: WMMA intra-chunk scan + sin + amp + store.
// D = L x F + C per 16-timestep tile:
//   A (16x4 f32, 2 VGPR): lanes 0-15 m=lane comps K={0,1}; lanes 16-31 K={2,3}
//   B (4x16 f32, 2 VGPR): lanes 0-15 n=lane comps K={0,1}; lanes 16-31 K={2,3}
//   C/D (16x16 f32, 8 VGPR): lanes 0-15 n=lane M=r; lanes 16-31 M=8+r
// Fast path is wave-uniform (tt0+16 <= T); only the last chunk's partial tile
// takes the guarded path, tiles fully past T break out.
// ---------------------------------------------------------------------------
__global__ void osc_main(const float* __restrict__ af,
                         const float* __restrict__ init_pos,
                         const float* __restrict__ init_ud,
                         const float* __restrict__ ws,
                         float* __restrict__ out) {
    int w     = blockIdx.x * WPB + (threadIdx.x >> 5);
    int lane  = threadIdx.x & 31;
    int chunk = w % NCHUNK;
    int rem   = w / NCHUNK;
    int cblk  = rem & 3;
    int b     = rem >> 2;
    int n     = lane & 15;
    int hi    = lane >> 4;
    int c     = cblk * 16 + n;
    int t0    = chunk * TCH;

    // theta0 in revolutions; every lane computes its own channel's value.
    const float inv2pi = 0.15915494309189535f;
    const float pi_f   = 3.14159265358979323846f;
    float p0 = init_pos[b * NC + c];
    p0 = fminf(1.f, fmaxf(-1.f, p0));
    float u0 = init_ud[b * NC + c];
    float a0 = asinf(p0);
    float th0_rev = ((u0 > 0.f) ? a0 : (pi_f - a0)) * inv2pi;

    // carry (chunk-exclusive sum of af1) + folded 2*theta0_rev, so that
    // phase_rev = 0.5 * D.
    float carry = ws[(chunk * NB + b) * NC + c] + 2.f * th0_rev;

    // Constant lower-triangular-ones A blocks: A_k[m, j] = (m >= 4k + j).
    int jbase = hi * 2;
    v2f A[4];
#pragma unroll
    for (int k = 0; k < 4; ++k) {
        A[k].x = (n >= 4 * k + jbase)     ? 1.f : 0.f;
        A[k].y = (n >= 4 * k + jbase + 1) ? 1.f : 0.f;
    }

    v8f C;
#pragma unroll
    for (int r = 0; r < 8; ++r) C[r] = carry;

    int m_base = hi << 3;   // D rows owned by this lane: m_base + r

    for (int tile = 0; tile < NTILES; ++tile) {
        int tt0 = t0 + tile * 16;
        if (tt0 >= T_TOTAL) break;          // rest of chunk is past the end

        v8f D = C;

        if (tt0 + 16 <= T_TOTAL) {
            // ---------------- fast path: branch-free, clustered loads ------
            const float* fbase = af + (b * T_TOTAL + tt0) * ROWF + c * 2 + 1;
            v2f Bv[4];
#pragma unroll
            for (int k = 0; k < 4; ++k) {
                Bv[k].x = fbase[(4 * k + jbase)     * ROWF];
                Bv[k].y = fbase[(4 * k + jbase + 1) * ROWF];
            }
#pragma unroll
            for (int k = 0; k < 4; ++k)
                D = __builtin_amdgcn_wmma_f32_16x16x4_f32(
                        false, A[k], false, Bv[k], (short)0, D, false, false);

            const float* abase = af + (b * T_TOTAL + tt0 + m_base) * ROWF + c * 2;
            float amp[8];
#pragma unroll
            for (int r = 0; r < 8; ++r)
                amp[r] = abase[r * ROWF];

            float* obase = out + (b * T_TOTAL + tt0 + m_base) * NC + c;
#pragma unroll
            for (int r = 0; r < 8; ++r) {
                float sv = __builtin_amdgcn_sinf(
                               __builtin_amdgcn_fractf(0.5f * D[r]));
                __builtin_nontemporal_store(sv * amp[r], &obase[r * NC]);
            }
        } else {
            // ---------------- tail tile (one per (b,cblk) in whole grid) ---
#pragma unroll
            for (int k = 0; k < 4; ++k) {
                int tA = tt0 + 4 * k + jbase;
                int tB = tA + 1;
                int tAc = (tA < T_TOTAL) ? tA : (T_TOTAL - 1);
                int tBc = (tB < T_TOTAL) ? tB : (T_TOTAL - 1);
                float fA = af[((b * T_TOTAL + tAc) * NC + c) * 2 + 1];
                float fB = af[((b * T_TOTAL + tBc) * NC + c) * 2 + 1];
                v2f Bv;
                Bv.x = (tA < T_TOTAL) ? fA : 0.f;
                Bv.y = (tB < T_TOTAL) ? fB : 0.f;
                D = __builtin_amdgcn_wmma_f32_16x16x4_f32(
                        false, A[k], false, Bv, (short)0, D, false, false);
            }
#pragma unroll
            for (int r = 0; r < 8; ++r) {
                int t = tt0 + m_base + r;
                if (t < T_TOTAL) {
                    float amp = af[((b * T_TOTAL + t) * NC + c) * 2 + 0];
                    float sv = __builtin_amdgcn_sinf(
                                   __builtin_amdgcn_fractf(0.5f * D[r]));
                    __builtin_nontemporal_store(
                        sv * amp, &out[(b * T_TOTAL + t) * NC + c]);
                }
            }
        }

        // Next tile's carry matrix = broadcast of D row 15 (lane 16+n, comp 7).
        float d15 = __int_as_float(
            __builtin_amdgcn_ds_bpermute((16 + n) << 2, __float_as_int(D[7])));
#pragma unroll
        for (int r = 0; r < 8; ++r) C[r] = d15;
    }
}

// ---------------------------------------------------------------------------
extern "C" void kernel_launch(void* const* d_in, const int* in_sizes, int n_in,
                              void* d_out, int out_size, void* d_ws, size_t ws_size,
                              hipStream_t stream) {
    const float* af = (const float*)d_in[0];   // (B,T,C,2) f32
    const float* ip = (const float*)d_in[1];   // (B,1,C,1) f32
    const float* iu = (const float*)d_in[2];   // (B,1,C,1) f32
    float* out = (float*)d_out;                // (B,T,C,1) f32
    float* ws  = (float*)d_ws;                 // NCHUNK*NB*NC*4 = ~406 KB

    int waves  = NB * 4 * NCHUNK;              // 6336
    int blocks = waves / WPB;                  // 792

    osc_partial_sums<<<blocks, 256, 0, stream>>>(af, ws);
    osc_scan_partials<<<(NB * NC + 255) / 256, 256, 0, stream>>>(ws);
    osc_main<<<blocks, 256, 0, stream>>>(af, ip, iu, ws, out);
}